// GCN_85452669321994
// MI455X (gfx1250) — compile-verified
//
#include <hip/hip_runtime.h>
#include <hip/hip_bf16.h>
#include <math.h>

#define HIDDEN 64
#define NGRAPH 512
#define NCLS   10

typedef __attribute__((ext_vector_type(2))) float v2f;
typedef __attribute__((ext_vector_type(8))) float v8f;

// ---------------- utility ----------------
__global__ void k_zero(float* __restrict__ p, long n) {
  long i = (long)blockIdx.x * blockDim.x + threadIdx.x;
  if (i < n) p[i] = 0.0f;
}

__global__ void k_init_deg(float* __restrict__ deg, int n) {
  int i = blockIdx.x * blockDim.x + threadIdx.x;
  if (i < n) deg[i] = 1.0f;   // self-loop contributes 1 to every node's degree
}

__global__ void k_deg_acc(const int* __restrict__ dst, float* __restrict__ deg, int e) {
  int i = blockIdx.x * blockDim.x + threadIdx.x;
  if (i < e) atomicAdd(&deg[dst[i]], 1.0f);
}

__global__ void k_dinv(const float* __restrict__ deg, float* __restrict__ dinv, int n) {
  int i = blockIdx.x * blockDim.x + threadIdx.x;
  if (i < n) dinv[i] = rsqrtf(deg[i]);   // deg >= 1 always (self-loops)
}

__global__ void k_norm(const int* __restrict__ src, const int* __restrict__ dst,
                       const float* __restrict__ dinv, float* __restrict__ norm, int e) {
  int i = blockIdx.x * blockDim.x + threadIdx.x;
  if (i < e) norm[i] = dinv[src[i]] * dinv[dst[i]];
}

// ---------------- GEMM: H[rows,64] = X[rows,64] @ W[64,64] via V_WMMA_F32_16X16X4_F32 ----------------
// One wave computes one 16x16 output tile; rows is a multiple of 16 (50000 = 3125*16).
__global__ void k_gemm64_wmma(const float* __restrict__ X, const float* __restrict__ W,
                              float* __restrict__ H, int rows) {
  int wave = (int)((blockIdx.x * blockDim.x + threadIdx.x) >> 5);
  int lane = threadIdx.x & 31;
  int tm = wave >> 2;          // row-tile (16 rows)
  int tn = wave & 3;           // col-tile (4 tiles of 16 over HIDDEN=64)
  int row0 = tm * 16;
  if (row0 >= rows) return;    // wave-uniform exit; EXEC all-1s otherwise

  int half = lane >> 4;        // 0: lanes 0-15, 1: lanes 16-31
  int l15  = lane & 15;
  int r = row0 + l15;          // A row for this lane
  int n = tn * 16 + l15;       // B/C column for this lane
  int koff = half * 2;         // K sub-offset per ISA fp32 A/B layout

  v8f c = {};
  #pragma unroll
  for (int kb = 0; kb < 16; ++kb) {       // K = 64 in steps of 4
    int k = kb * 4 + koff;
    v2f a, b;
    a.x = X[(long)r * HIDDEN + k];
    a.y = X[(long)r * HIDDEN + k + 1];
    b.x = W[(long)k * HIDDEN + n];
    b.y = W[(long)(k + 1) * HIDDEN + n];
    c = __builtin_amdgcn_wmma_f32_16x16x4_f32(false, a, false, b, (short)0, c,
                                              false, false);
  }
  // C/D layout: VGPR v -> row = row0 + v + 8*half, col = n
  int rbase = row0 + half * 8;
  #pragma unroll
  for (int v = 0; v < 8; ++v)
    H[(long)(rbase + v) * HIDDEN + n] = c[v];
}

// ---------------- edge scatter: AGG[dst] += H[src] * norm ----------------
// 16 threads per edge, float4 per thread.
__global__ void k_scatter(const float* __restrict__ H, const int* __restrict__ src,
                          const int* __restrict__ dst, const float* __restrict__ norm,
                          float* __restrict__ AGG, int e) {
  long t = (long)blockIdx.x * blockDim.x + threadIdx.x;
  int edge = (int)(t >> 4);
  if (edge >= e) return;
  int f = ((int)t & 15) * 4;
  int s = src[edge], d = dst[edge];
  float w = norm[edge];
  const float4 v = *reinterpret_cast<const float4*>(H + (long)s * HIDDEN + f);
  float* out = AGG + (long)d * HIDDEN + f;
  atomicAdd(out + 0, v.x * w);
  atomicAdd(out + 1, v.y * w);
  atomicAdd(out + 2, v.z * w);
  atomicAdd(out + 3, v.w * w);
}

// ---------------- finalize: feat = relu(agg + h*dinv^2 + bias) (in place on AGG) ----------------
__global__ void k_finalize(const float* __restrict__ H, float* __restrict__ AGG,
                           const float* __restrict__ dinv, const float* __restrict__ bias,
                           int n) {
  long idx = (long)blockIdx.x * blockDim.x + threadIdx.x;
  if (idx >= (long)n * HIDDEN) return;
  int node = (int)(idx >> 6);
  int f = (int)(idx & 63);
  float di = dinv[node];
  float v = AGG[idx] + H[idx] * di * di + bias[f];
  AGG[idx] = fmaxf(v, 0.0f);
}

// ---------------- mean pool: sums[g] += feat[node], cnts[g] += 1 ----------------
__global__ void k_pool(const float* __restrict__ feat, const int* __restrict__ batch,
                       float* __restrict__ sums, float* __restrict__ cnts, int n) {
  long t = (long)blockIdx.x * blockDim.x + threadIdx.x;
  int node = (int)(t >> 4);
  if (node >= n) return;
  int f = ((int)t & 15) * 4;
  int g = batch[node];
  const float4 v = *reinterpret_cast<const float4*>(feat + (long)node * HIDDEN + f);
  float* out = sums + (long)g * HIDDEN + f;
  atomicAdd(out + 0, v.x);
  atomicAdd(out + 1, v.y);
  atomicAdd(out + 2, v.z);
  atomicAdd(out + 3, v.w);
  if (f == 0) atomicAdd(&cnts[g], 1.0f);
}

// ---------------- head: g = sums/cnt; relu(g@lin1+b1); g@lin2+b2; log_softmax ----------------
__global__ void k_head(const float* __restrict__ sums, const float* __restrict__ cnts,
                       const float* __restrict__ w1, const float* __restrict__ b1,
                       const float* __restrict__ w2, const float* __restrict__ b2,
                       float* __restrict__ out) {
  int g = blockIdx.x * blockDim.x + threadIdx.x;
  if (g >= NGRAPH) return;
  float inv = 1.0f / fmaxf(cnts[g], 1.0f);
  float gv[HIDDEN];
  #pragma unroll
  for (int k = 0; k < HIDDEN; ++k) gv[k] = sums[(long)g * HIDDEN + k] * inv;
  float hid[HIDDEN];
  for (int j = 0; j < HIDDEN; ++j) {
    float acc = b1[j];
    #pragma unroll
    for (int k = 0; k < HIDDEN; ++k) acc = fmaf(gv[k], w1[k * HIDDEN + j], acc);
    hid[j] = fmaxf(acc, 0.0f);
  }
  float logits[NCLS];
  float mx = -INFINITY;
  for (int c = 0; c < NCLS; ++c) {
    float acc = b2[c];
    #pragma unroll
    for (int j = 0; j < HIDDEN; ++j) acc = fmaf(hid[j], w2[j * NCLS + c], acc);
    logits[c] = acc;
    mx = fmaxf(mx, acc);
  }
  float se = 0.0f;
  for (int c = 0; c < NCLS; ++c) se += expf(logits[c] - mx);
  float lse = mx + logf(se);
  for (int c = 0; c < NCLS; ++c) out[(long)g * NCLS + c] = logits[c] - lse;
}

// ---------------- launch ----------------
extern "C" void kernel_launch(void* const* d_in, const int* in_sizes, int n_in,
                              void* d_out, int out_size, void* d_ws, size_t ws_size,
                              hipStream_t stream) {
  const float* x      = (const float*)d_in[0];
  const int*   eidx   = (const int*)d_in[1];
  const int*   batch  = (const int*)d_in[2];
  const float* W1     = (const float*)d_in[3];
  const float* b1     = (const float*)d_in[4];
  const float* W2     = (const float*)d_in[5];
  const float* b2     = (const float*)d_in[6];
  const float* W3     = (const float*)d_in[7];
  const float* b3     = (const float*)d_in[8];
  const float* lin1_w = (const float*)d_in[9];
  const float* lin1_b = (const float*)d_in[10];
  const float* lin2_w = (const float*)d_in[11];
  const float* lin2_b = (const float*)d_in[12];

  const int N = in_sizes[0] / HIDDEN;      // 50000
  const int E = in_sizes[1] / 2;           // 1200000
  const int* src = eidx;                   // edge_index[0]
  const int* dst = eidx + E;               // edge_index[1]

  // workspace layout (floats)
  float* ws   = (float*)d_ws;
  float* bufH = ws;                        // N*64 : GEMM output
  float* bufA = bufH + (long)N * HIDDEN;   // N*64 : aggregation / layer output (in place)
  float* deg  = bufA + (long)N * HIDDEN;   // N
  float* dinv = deg + N;                   // N
  float* norm = dinv + N;                  // E
  float* sums = norm + E;                  // 512*64
  float* cnts = sums + (long)NGRAPH * HIDDEN; // 512
  (void)ws_size; (void)n_in; (void)out_size;

  const int TB = 256;
  auto blocks = [](long n, int tb) { return (int)((n + tb - 1) / tb); };

  // ---- normalization coefficients (computed once, reused by all 3 layers) ----
  k_init_deg<<<blocks(N, TB), TB, 0, stream>>>(deg, N);
  k_deg_acc<<<blocks(E, TB), TB, 0, stream>>>(dst, deg, E);
  k_dinv<<<blocks(N, TB), TB, 0, stream>>>(deg, dinv, N);
  k_norm<<<blocks(E, TB), TB, 0, stream>>>(src, dst, dinv, norm, E);

  // ---- 3 GCN layers ----
  const float* Wl[3] = {W1, W2, W3};
  const float* bl[3] = {b1, b2, b3};
  const float* in = x;
  const long nWaves = ((long)(N + 15) / 16) * 4;    // 16x16 tiles
  const long gemmThreads = nWaves * 32;
  const long scatThreads = (long)E * 16;
  const long elemThreads = (long)N * HIDDEN;

  for (int l = 0; l < 3; ++l) {
    k_gemm64_wmma<<<blocks(gemmThreads, TB), TB, 0, stream>>>(in, Wl[l], bufH, N);
    k_zero<<<blocks(elemThreads, TB), TB, 0, stream>>>(bufA, elemThreads);
    k_scatter<<<blocks(scatThreads, TB), TB, 0, stream>>>(bufH, src, dst, norm, bufA, E);
    k_finalize<<<blocks(elemThreads, TB), TB, 0, stream>>>(bufH, bufA, dinv, bl[l], N);
    in = bufA;
  }

  // ---- global mean pool ----
  k_zero<<<blocks((long)NGRAPH * HIDDEN + NGRAPH, TB), TB, 0, stream>>>(
      sums, (long)NGRAPH * HIDDEN + NGRAPH);   // sums and cnts are contiguous
  k_pool<<<blocks((long)N * 16, TB), TB, 0, stream>>>(bufA, batch, sums, cnts, N);

  // ---- MLP head + log_softmax ----
  k_head<<<blocks(NGRAPH, TB), TB, 0, stream>>>(sums, cnts, lin1_w, lin1_b,
                                                lin2_w, lin2_b, (float*)d_out);
}